// RelativeSpacetimeBias_30210799960114
// MI455X (gfx1250) — compile-verified
//
#include <hip/hip_runtime.h>
#include <hip/hip_bf16.h>

// RelativeSpacetimeBias on gfx1250 (MI455X).
//
// ds2(i,j) = Q_i + Q_j - 2 * sum_c g_c p_ic p_jc,  g = (-1,-1,-1,+1),
//            Q = sum_c g_c p_c^2
// -> two V_WMMA_F32_16X16X4_F32 per 16x16 output tile produce the full ds2
//    tile in the f32 accumulator (first WMMA: rank-4 cross term with the
//    -2*g metric folded into A; second WMMA: rank-2 [Q_i,1] x [1,Q_j]^T
//    accumulated into the same C).
// -> |ds| <= 2^-8 after clip/scale, so sin/cos come from a Taylor seed +
//    double-angle recurrence: no transcendentals except raw v_sqrt_f32
//    (argument >= 1e-12, far from denorms -> skip the precise-sqrt fixup).
// Store-bound: 67 MB out @ 23.3 TB/s ~= 2.9 us floor.

#define NUM_FREQ 16
#define SDIM     2048
#define BDIM     4
#define TILES_PER_ROW   (SDIM / 16)                 // 128
#define TILES_PER_BATCH (TILES_PER_ROW * TILES_PER_ROW)  // 16384
#define TOTAL_TILES     (BDIM * TILES_PER_BATCH)    // 65536
#define WAVES_PER_BLOCK 8

typedef __attribute__((ext_vector_type(2))) float v2f;
typedef __attribute__((ext_vector_type(8))) float v8f;

__global__ __launch_bounds__(256)
void RelativeSpacetimeBias_30210799960114_kernel(
    const float* __restrict__ pos,      // (B, S, 4)
    const int*   __restrict__ vmask,    // (B, S)
    const float* __restrict__ proj_w,   // (1, 32)
    const float* __restrict__ proj_b,   // (1,)
    float*       __restrict__ out)      // (B, S, S)
{
    const int lane = threadIdx.x & 31;
    const int wave = threadIdx.x >> 5;
    const int tile = blockIdx.x * WAVES_PER_BLOCK + wave;

    const int b   = tile >> 14;          // tile / TILES_PER_BATCH
    const int rem = tile & (TILES_PER_BATCH - 1);
    const int i0  = (rem >> 7) << 4;     // tile-row * 16
    const int j0  = (rem & (TILES_PER_ROW - 1)) << 4;  // tile-col * 16

    const int half = lane >> 4;          // 0: K=0,1 ; 1: K=2,3
    const int li   = lane & 15;          // row of A / col of B for this lane

    // --- load the two 16x4 position panels for this tile -------------------
    const float* pbase = pos + (size_t)b * SDIM * 4;
    const float4 pi = *(const float4*)(pbase + (size_t)(i0 + li) * 4);
    const float4 pj = *(const float4*)(pbase + (size_t)(j0 + li) * 4);

    // Q = g-weighted squared norm (metric -,-,-,+ on x,y,z,t)
    const float Qi = pi.w * pi.w - pi.x * pi.x - pi.y * pi.y - pi.z * pi.z;
    const float Qj = pj.w * pj.w - pj.x * pj.x - pj.y * pj.y - pj.z * pj.z;

    // WMMA #1 operands: A[m][c] = -2*g_c*p_i[m][c], B[c][n] = p_j[n][c].
    // A-layout (16x4 f32): lanes 0-15 -> K=0,1 ; lanes 16-31 -> K=2,3.
    // B-layout (4x16 f32): mirrored (lane = column).
    v2f a1, b1, a2, b2;
    a1[0] = half ?  2.0f * pi.z :  2.0f * pi.x;   // -2*g: x,y,z -> +2 ; t -> -2
    a1[1] = half ? -2.0f * pi.w :  2.0f * pi.y;
    b1[0] = half ?         pj.z :         pj.x;
    b1[1] = half ?         pj.w :         pj.y;
    // WMMA #2 operands: A2[m] = [Q_i, 1, 0, 0], B2[:,n] = [1, Q_j, 0, 0]
    a2[0] = half ? 0.0f : Qi;
    a2[1] = half ? 0.0f : 1.0f;
    b2[0] = half ? 0.0f : 1.0f;
    b2[1] = half ? 0.0f : Qj;

    v8f c = {};
    c = __builtin_amdgcn_wmma_f32_16x16x4_f32(false, a1, false, b1,
                                              (short)0, c, false, false);
    c = __builtin_amdgcn_wmma_f32_16x16x4_f32(false, a2, false, b2,
                                              (short)0, c, false, false);
    // c[r] now holds ds2 for (row = r + 8*half, col = li) of this tile.

    // --- projection weights (uniform, L2-resident) --------------------------
    float ws[2 * NUM_FREQ];
#pragma unroll
    for (int f = 0; f < 2 * NUM_FREQ; ++f) ws[f] = proj_w[f];
    const float bias0 = proj_b[0];

    const int* vb = vmask + (size_t)b * SDIM;
    const int  mj = vb[j0 + li];

    float* obase = out + (size_t)b * SDIM * SDIM;

#pragma unroll
    for (int r = 0; r < 8; ++r) {
        const int row = r + 8 * half;
        const float ds2 = c[r];

        // ds = sign(ds2) * sqrt(|ds2| + 1e-12), clipped, scaled.
        // Raw v_sqrt_f32: argument >= 1e-12, no denorm rescale needed.
        float mag = __builtin_amdgcn_sqrtf(__builtin_fabsf(ds2) + 1e-12f);
        float ds  = (ds2 > 0.0f) ? mag : ((ds2 < 0.0f) ? -mag : 0.0f);
        ds = fminf(fmaxf(ds, -4.0f), 4.0f) * (1.0f / 1024.0f);

        // |ds| <= 2^-8: Taylor seed exact to f32 ulp
        const float x2 = ds * ds;
        float s  = ds * (1.0f - (1.0f / 6.0f) * x2);
        float cc = 1.0f - 0.5f * x2 + (1.0f / 24.0f) * (x2 * x2);

        // sum_f sin(ds*2^f)*w[f] + cos(ds*2^f)*w[16+f] via double-angle
        float acc = bias0;
#pragma unroll
        for (int f = 0; f < NUM_FREQ; ++f) {
            acc = fmaf(s,  ws[f],            acc);
            acc = fmaf(cc, ws[NUM_FREQ + f], acc);
            const float ns = 2.0f * s * cc;           // sin(2x)
            cc = fmaf(cc, cc, -(s * s));              // cos(2x) = c^2 - s^2
            s  = ns;
        }

        const int   mi   = vb[i0 + row];
        const float outv = (mi && mj) ? acc : 0.0f;
        obase[(size_t)(i0 + row) * SDIM + (j0 + li)] = outv;
    }
}

extern "C" void kernel_launch(void* const* d_in, const int* in_sizes, int n_in,
                              void* d_out, int out_size, void* d_ws, size_t ws_size,
                              hipStream_t stream) {
    (void)in_sizes; (void)n_in; (void)d_ws; (void)ws_size; (void)out_size;
    const float* pos    = (const float*)d_in[0];
    const int*   vmask  = (const int*)d_in[1];
    const float* proj_w = (const float*)d_in[2];
    const float* proj_b = (const float*)d_in[3];
    float*       out    = (float*)d_out;

    const int blocks = TOTAL_TILES / WAVES_PER_BLOCK;  // 8192
    RelativeSpacetimeBias_30210799960114_kernel<<<blocks, 256, 0, stream>>>(
        pos, vmask, proj_w, proj_b, out);
}